// SiegelmannSontag1_60198261621091
// MI455X (gfx1250) — compile-verified
//
#include <hip/hip_runtime.h>

// SiegelmannSontag step, fused. Bandwidth-bound: ~183 MB weights / 23.3 TB/s ~ 8us.
// Strategy: one wave handles 16-row tiles of the n=314928 dimension.
//   Stage 1: cd16 = sat(Wcd[R:R+16, 0:80] @ w_in + b_cd)  via 20x V_WMMA_F32_16X16X4_F32
//            (B = w_in chunk broadcast over all 16 N columns -> every D column identical)
//   Stage 2: persistent acc64 += [Wbeta;Wgamma][:, R:R+16] @ cd16 via 16x WMMA per tile
// Per-wave partials -> per-block partial (LDS, fixed order) -> workspace
// -> deterministic final reduction + 16-dim affine tail in a second kernel.

typedef __attribute__((ext_vector_type(2))) float v2f;
typedef __attribute__((ext_vector_type(8))) float v8f;

#define N_ROWS 314928   // 48 * 9^4
#define KDIM   80       // 8*p + s = 32 + 48
#define NTILES (N_ROWS / 16)  // 19683

__device__ __forceinline__ float satf(float v) {
    return fminf(fmaxf(v, 0.0f), 1.0f);
}

__global__ __launch_bounds__(256) void ss_main_kernel(
    const float* __restrict__ x,
    const float* __restrict__ Wcd,
    const float* __restrict__ bcd,
    const float* __restrict__ Wbeta,
    const float* __restrict__ Wgamma,
    float* __restrict__ partials,     // [gridDim.x * 64]
    int nWavesTotal)
{
    __shared__ float red[8 * 64];

    const int lane = threadIdx.x & 31;
    const int wid  = threadIdx.x >> 5;
    const int hi   = lane >> 4;       // which lane-half (selects K pair / M+8)
    const int l15  = lane & 15;
    const int gw   = blockIdx.x * 8 + wid;

    // ---- preload w_in pairs (tile-invariant). w_in[k] = (k<32) ? x[64+k] : x[k-32]
    // lane needs {w[4c+2*hi], w[4c+2*hi+1]} for each K-chunk c (B-fragment, broadcast over N).
    v2f wv[20];
#pragma unroll
    for (int c = 0; c < 20; ++c) {
        const float* base = (c < 8) ? (x + 64) : (x - 32);  // boundary k=32 at c=8, both pair elems same side
        const int k0 = 4 * c + 2 * hi;
        wv[c] = (v2f){ base[k0], base[k0 + 1] };
    }

    // ---- per-lane row pointers into the virtual 64-row [Wbeta(48); Wgamma(16)] matrix
    const float* mrow[4];
#pragma unroll
    for (int mt = 0; mt < 4; ++mt) {
        const int r = 16 * mt + l15;
        mrow[mt] = (mt < 3) ? (Wbeta + (size_t)r * N_ROWS)
                            : (Wgamma + (size_t)l15 * N_ROWS);
    }

    const v8f vzero = {0.f, 0.f, 0.f, 0.f, 0.f, 0.f, 0.f, 0.f};
    v8f accM[4];
#pragma unroll
    for (int mt = 0; mt < 4; ++mt) accM[mt] = vzero;

    for (int t = gw; t < NTILES; t += nWavesTotal) {
        const int R = t * 16;

        // ---------- Stage 1: cd = sat(Wcd_tile @ w_in + b) ----------
        // A fragment (documented layout): lanes 0-15 hold rows R+l15 with K = {4c, 4c+1},
        // lanes 16-31 hold rows R+l15 with K = {4c+2, 4c+3}.
        const float* arow = Wcd + (size_t)(R + l15) * KDIM + 2 * hi;
        // prefetch this lane's row for the NEXT tile we will process
        __builtin_prefetch(arow + (size_t)nWavesTotal * 16 * KDIM, 0, 1);

        v8f acc = vzero;
#pragma unroll
        for (int c = 0; c < 20; ++c) {
            v2f a = *(const v2f*)(arow + 4 * c);
            acc = __builtin_amdgcn_wmma_f32_16x16x4_f32(
                false, a, false, wv[c], (short)0, acc, false, false);
        }

        // C/D layout: acc[j] holds row M = j + 8*hi (all N columns identical by construction)
        const float* bp = bcd + R + 8 * hi;
        float cdh[8];
#pragma unroll
        for (int j = 0; j < 8; ++j) cdh[j] = satf(acc[j] + bp[j]);

        // redistribute: every lane gets full cd[0..15] (sources: lane 0 for M<8, lane 16 for M>=8)
        float cd[16];
#pragma unroll
        for (int m = 0; m < 16; ++m)
            cd[m] = __shfl(cdh[m & 7], (m >> 3) << 4, 32);

        // ---------- Stage 2: acc64 += [Wbeta;Wgamma]_tile(64x16) @ cd(16) ----------
#pragma unroll
        for (int mt = 0; mt < 4; ++mt) {
            const float* mr = mrow[mt] + R + 2 * hi;
#pragma unroll
            for (int c = 0; c < 4; ++c) {
                v2f a = *(const v2f*)(mr + 4 * c);
                v2f b;
                b.x = hi ? cd[4 * c + 2] : cd[4 * c + 0];
                b.y = hi ? cd[4 * c + 3] : cd[4 * c + 1];
                accM[mt] = __builtin_amdgcn_wmma_f32_16x16x4_f32(
                    false, a, false, b, (short)0, accM[mt], false, false);
            }
        }
    }

    // ---- per-wave -> LDS: out64[16*mt + 8*hi + j] = accM[mt][j] (lanes 0 and 16 hold column 0)
    if (l15 == 0) {
#pragma unroll
        for (int mt = 0; mt < 4; ++mt)
#pragma unroll
            for (int j = 0; j < 8; ++j)
                red[wid * 64 + 16 * mt + 8 * hi + j] = accM[mt][j];
    }
    __syncthreads();

    // fixed-order block reduction (deterministic)
    if (threadIdx.x < 64) {
        float s = 0.f;
#pragma unroll
        for (int w = 0; w < 8; ++w) s += red[w * 64 + threadIdx.x];
        partials[(size_t)blockIdx.x * 64 + threadIdx.x] = s;
    }
}

__global__ __launch_bounds__(64) void ss_final_kernel(
    const float* __restrict__ x,
    const float* __restrict__ Wstack,
    const float* __restrict__ Wnstack, const float* __restrict__ bnstack,
    const float* __restrict__ Wntop,
    const float* __restrict__ Wsubtop, const float* __restrict__ bsubtop,
    const float* __restrict__ Wsubne,  const float* __restrict__ bsubne,
    const float* __restrict__ partials, int nblk,
    float* __restrict__ out)
{
    __shared__ float beta64[64];
    __shared__ float nns[16];
    const int tid = threadIdx.x;

    // deterministic fixed-order reduction over block partials
    float s = 0.f;
    for (int b = 0; b < nblk; ++b) s += partials[(size_t)b * 64 + tid];
    beta64[tid] = s;
    if (tid < 48) out[tid] = s;           // next_state = Wbeta @ cd_out
    __syncthreads();

    if (tid < 16) {
        // stack = Wstack @ sat(x[48:64]);  top = Wstack @ sat(x[64:80])
        float st[4], tp[4];
#pragma unroll
        for (int t = 0; t < 4; ++t) {
            float a = 0.f, c = 0.f;
            for (int j = 0; j < 16; ++j) {
                a += Wstack[t * 16 + j] * satf(x[48 + j]);
                c += Wstack[t * 16 + j] * satf(x[64 + j]);
            }
            st[t] = a; tp[t] = c;
        }
        float v = bnstack[tid] + beta64[48 + tid] - 1.0f;  // + Wgamma@cd_out - 1
#pragma unroll
        for (int t = 0; t < 4; ++t)
            v += Wnstack[tid * 4 + t] * st[t] + Wntop[tid * 4 + t] * tp[t];
        out[48 + tid] = v;
        nns[tid] = v;
    }
    __syncthreads();

    if (tid < 16) {
        float a = bsubtop[tid], c = bsubne[tid];
        for (int j = 0; j < 16; ++j) {
            a += Wsubtop[tid * 16 + j] * nns[j];
            c += Wsubne[tid * 16 + j]  * nns[j];
        }
        out[64 + tid] = a;   // next_noisy_sub_top
        out[80 + tid] = c;   // next_noisy_sub_nonempty
    }
}

extern "C" void kernel_launch(void* const* d_in, const int* in_sizes, int n_in,
                              void* d_out, int out_size, void* d_ws, size_t ws_size,
                              hipStream_t stream) {
    (void)in_sizes; (void)n_in; (void)out_size;
    const float* x       = (const float*)d_in[0];
    const float* Wcd     = (const float*)d_in[1];
    const float* bcd     = (const float*)d_in[2];
    const float* Wstack  = (const float*)d_in[3];
    const float* Wbeta   = (const float*)d_in[4];
    const float* Wgamma  = (const float*)d_in[5];
    const float* Wnstack = (const float*)d_in[6];
    const float* bnstack = (const float*)d_in[7];
    const float* Wntop   = (const float*)d_in[8];
    const float* Wsubtop = (const float*)d_in[9];
    const float* bsubtop = (const float*)d_in[10];
    const float* Wsubne  = (const float*)d_in[11];
    const float* bsubne  = (const float*)d_in[12];
    float* out      = (float*)d_out;
    float* partials = (float*)d_ws;

    int nblk = 256;  // 2048 waves, ~10 tiles each
    const size_t need = (size_t)nblk * 64 * sizeof(float);
    if (ws_size < need) {
        nblk = (int)(ws_size / (64 * sizeof(float)));
        if (nblk < 1) nblk = 1;
    }

    ss_main_kernel<<<nblk, 256, 0, stream>>>(x, Wcd, bcd, Wbeta, Wgamma,
                                             partials, nblk * 8);
    ss_final_kernel<<<1, 64, 0, stream>>>(x, Wstack, Wnstack, bnstack, Wntop,
                                          Wsubtop, bsubtop, Wsubne, bsubne,
                                          partials, nblk, out);
}